// MultiHeadAttentionDisturbLayer_62706522521849
// MI455X (gfx1250) — compile-verified
//
#include <hip/hip_runtime.h>
#include <hip/hip_bf16.h>

// MI455X / gfx1250, wave32. All matrix math via V_WMMA_F32_16X16X4_F32.
//
// Refactored algorithm (mathematically identical to reference):
//   q_proj = query@Wq + bq
//   t[b,h,m]   = sum_c Wk[m, h*64+c] * q_proj[b, h*64+c]
//   sbias[b,h] = sum_c bk[h*64+c]    * q_proj[b, h*64+c]
//   scores[b,h,n] = key[b,n,:] . t[b,h,:] + sbias[b,h]   (disturb h=0,3 -> 1)
//   w = softmax(scores / 32)
//   u[b,h,m] = sum_n w[b,h,n] * value[b,n,m]
//   out[b,h*64+d] = u[b,h,:] . Wv[:, h*64+d] + bv[h*64+d]
// Total FLOPs ~16G (vs ~520G naive) -> HBM-bound on key/value (~1 GB, ~47us
// roofline at 23.3 TB/s). k_attn is a double-buffered software pipeline:
// chunk c+1's global->LDS traffic is in flight while WMMAs consume chunk c.
// The toolchain exposes the CDNA5 async-to-LDS builtins (confirmed by probe),
// so staging goes through the async engine (ASYNCcnt); register-prefetch
// fallback kept for portability.

#define NUM_HEADS 16
#define HEAD_DIM 64
#define MODEL_DIM 1024
#define N_TOK 121
#define NPAD 128
#define BATCH 1024
#define MC 64                 // m-chunk width staged per pipeline stage
#define TSTR 76               // LDS row stride (bank-conflict free for A & B frags)
#define TILE_F (NPAD * TSTR)

#if defined(__has_builtin)
#if __has_builtin(__builtin_amdgcn_global_load_async_to_lds_b128) && \
    __has_builtin(__builtin_amdgcn_s_wait_asynccnt)
#define USE_ASYNC_LDS 1
#endif
#endif
#ifndef USE_ASYNC_LDS
#define USE_ASYNC_LDS 0
#endif

typedef float v2f __attribute__((ext_vector_type(2)));
typedef float v8f __attribute__((ext_vector_type(8)));
#if USE_ASYNC_LDS
// builtin signature (from compiler diagnostic): param0 = int4-vector pointer
// in the global (__device__/AS1) address space, non-const.
typedef int ai4 __attribute__((ext_vector_type(4)));
typedef __attribute__((address_space(1))) ai4 gai4;   // global int4
typedef __attribute__((address_space(3))) ai4 lai4;   // LDS int4
#endif

// ---------------- generic fp32 WMMA 16x16 tile (K multiple of 4) -------------
// A fragment (16x4 f32): lane L -> row L&15, K-pair 2*(L>>4): vgpr0/1 = K,K+1.
// B fragment (4x16 f32): lane L -> col L&15, K rows 2*(L>>4)+{0,1}.
// C/D: vgpr j -> row j + 8*(L>>4), col L&15.
__device__ __forceinline__ v8f wmma_gemm_tile(
    const float* __restrict__ A, int sAm,
    const float* __restrict__ B, int sBk, int sBn,
    int K, int lane) {
  v8f acc = {0.f,0.f,0.f,0.f,0.f,0.f,0.f,0.f};
  const int r  = lane & 15;
  const int kk = (lane >> 4) << 1;
  for (int k = 0; k < K; k += 4) {
    v2f a, b;
    a.x = A[r * sAm + (k + kk)];
    a.y = A[r * sAm + (k + kk + 1)];
    b.x = B[(k + kk) * sBk + r * sBn];
    b.y = B[(k + kk + 1) * sBk + r * sBn];
    acc = __builtin_amdgcn_wmma_f32_16x16x4_f32(
        false, a, false, b, (short)0, acc, false, false);
  }
  return acc;
}

__device__ __forceinline__ void store_tile(float* __restrict__ C, int sCm,
                                           const v8f& acc,
                                           const float* __restrict__ bias,
                                           int lane) {
  const int r  = lane & 15;
  const int mh = (lane >> 4) << 3;
  const float bv = bias ? bias[r] : 0.0f;
#pragma unroll
  for (int j = 0; j < 8; ++j)
    C[(j + mh) * sCm + r] = acc[j] + bv;
}

// -------------------------- K1: q_proj = query@Wq + bq -----------------------
__global__ void k_qproj(const float* __restrict__ query,
                        const float* __restrict__ Wq,
                        const float* __restrict__ bq,
                        float* __restrict__ qproj) {
  const int wave = (blockIdx.x * blockDim.x + threadIdx.x) >> 5;
  const int lane = threadIdx.x & 31;
  const int tm = (wave >> 6) << 4;
  const int tn = (wave & 63) << 4;
  v8f acc = wmma_gemm_tile(query + tm * MODEL_DIM, MODEL_DIM,
                           Wq + tn, MODEL_DIM, 1, MODEL_DIM, lane);
  store_tile(qproj + tm * MODEL_DIM + tn, MODEL_DIM, acc, bq + tn, lane);
}

// ----------------- K2a: t[b,h,m] = q_h[b,:] @ Wk_h^T  (K=64) -----------------
__global__ void k_tproj(const float* __restrict__ qproj,
                        const float* __restrict__ Wk,
                        float* __restrict__ t) {
  const int wave = (blockIdx.x * blockDim.x + threadIdx.x) >> 5;
  const int lane = threadIdx.x & 31;
  const int h   = wave >> 12;
  const int rem = wave & 4095;
  const int tb = (rem >> 6) << 4;
  const int tm = (rem & 63) << 4;
  v8f acc = wmma_gemm_tile(qproj + tb * MODEL_DIM + h * HEAD_DIM, MODEL_DIM,
                           Wk + h * HEAD_DIM + tm * MODEL_DIM, 1, MODEL_DIM,
                           HEAD_DIM, lane);
  store_tile(t + (tb * NUM_HEADS + h) * MODEL_DIM + tm,
             NUM_HEADS * MODEL_DIM, acc, nullptr, lane);
}

// ------------------------ K2b: sbias[b,h] = q_h . bk_h -----------------------
__global__ void k_sbias(const float* __restrict__ qproj,
                        const float* __restrict__ bk,
                        float* __restrict__ sb) {
  const int idx = blockIdx.x * blockDim.x + threadIdx.x;
  if (idx >= BATCH * NUM_HEADS) return;
  const int h = idx & 15, b = idx >> 4;
  float s = 0.f;
#pragma unroll 8
  for (int c = 0; c < HEAD_DIM; ++c)
    s += qproj[b * MODEL_DIM + h * HEAD_DIM + c] * bk[h * HEAD_DIM + c];
  sb[idx] = s;
}

// --------------- K3: fused scores -> softmax -> u  (1 block / batch) ---------
__global__ __launch_bounds__(256)
void k_attn(const float* __restrict__ key, const float* __restrict__ value,
            const float* __restrict__ t, const float* __restrict__ sb,
            float* __restrict__ u, float* __restrict__ wout) {
  __shared__ float tile[2 * TILE_F];        // key/value chunks [n][m], stride 76
  __shared__ float tTc[2][MC * 17];         // t chunk transposed [m][h], padded
  __shared__ float S[NUM_HEADS * 129];      // scores [h][n], padded
  __shared__ float wT[NPAD * NUM_HEADS];    // weights transposed [n][h]
  __shared__ float sbs[NUM_HEADS];

  const int b    = blockIdx.x;
  const int tid  = threadIdx.x;
  const int lane = tid & 31;
  const int w    = tid >> 5;                // 8 waves
  const int r    = lane & 15;
  const int kk   = (lane >> 4) << 1;
  const int mh   = (lane >> 4) << 3;
  const int n0   = w << 4;

  const float* kb = key   + (long)b * N_TOK * MODEL_DIM;
  const float* vb = value + (long)b * N_TOK * MODEL_DIM;
  const float* tg = t     + (long)b * NUM_HEADS * MODEL_DIM;

  // one-time zero (pad rows n>=121 must read as 0; async path relies on this)
  for (int i = tid; i < 2 * TILE_F; i += 256) tile[i] = 0.f;
  if (tid < NUM_HEADS) sbs[tid] = sb[b * NUM_HEADS + tid];

  // ---- staging helpers: slot j -> row n = (tid+256j)>>4, float4 col (..)&15 -
#if USE_ASYNC_LDS
  auto issue_tile = [&](const float* base, int mc, int bufsel) {
    float* dst = tile + bufsel * TILE_F;
#pragma unroll
    for (int j = 0; j < 8; ++j) {
      const int idx = tid + (j << 8);
      const int n = idx >> 4, m4 = idx & 15;
      if (n < N_TOK)   // EXEC-masked lanes leave pre-zeroed pad rows intact
        __builtin_amdgcn_global_load_async_to_lds_b128(
            (gai4*)(base + n * MODEL_DIM + mc + (m4 << 2)),
            (lai4*)(dst + n * TSTR + (m4 << 2)),
            0, 0);
    }
  };
#else
  float4 pf[8];
  auto issue_tile = [&](const float* base, int mc, int /*bufsel*/) {
#pragma unroll
    for (int j = 0; j < 8; ++j) {
      const int idx = tid + (j << 8);
      const int n = idx >> 4, m4 = idx & 15;
      pf[j] = (n < N_TOK)
                  ? *(const float4*)(base + n * MODEL_DIM + mc + (m4 << 2))
                  : make_float4(0.f, 0.f, 0.f, 0.f);
    }
  };
  auto commit_tile = [&](int bufsel) {
    float* dst = tile + bufsel * TILE_F;
#pragma unroll
    for (int j = 0; j < 8; ++j) {
      const int idx = tid + (j << 8);
      *(float4*)(dst + (idx >> 4) * TSTR + ((idx & 15) << 2)) = pf[j];
    }
  };
#endif
  float tpf[4];
  auto issue_t = [&](int mc) {               // coalesced read over [h][ml]
#pragma unroll
    for (int j = 0; j < 4; ++j) {
      const int idx = tid + (j << 8);
      tpf[j] = tg[(idx >> 6) * MODEL_DIM + mc + (idx & 63)];
    }
  };
  auto commit_t = [&](int bufsel) {
#pragma unroll
    for (int j = 0; j < 4; ++j) {
      const int idx = tid + (j << 8);
      tTc[bufsel][(idx & 63) * 17 + (idx >> 6)] = tpf[j];
    }
  };

  __syncthreads();                          // zero-init visible to all
  issue_t(0);
  issue_tile(kb, 0, 0);

  // ---- scores pipeline: S[n,h] += key_chunk . t_chunk, 16 chunks of 64 ------
  v8f acc = {0.f,0.f,0.f,0.f,0.f,0.f,0.f,0.f};
  for (int c = 0; c < 16; ++c) {
    const int cur = c & 1;
#if USE_ASYNC_LDS
    __builtin_amdgcn_s_wait_asynccnt(0);
#else
    commit_tile(cur);
#endif
    commit_t(cur);
    __syncthreads();   // single barrier/iter: double buffering covers the rest
    if (c < 15) { issue_t((c + 1) * MC); issue_tile(kb, (c + 1) * MC, cur ^ 1); }
    const float* buf = tile + cur * TILE_F;
#pragma unroll 8
    for (int k = 0; k < MC; k += 4) {
      v2f a, bb;
      a.x  = buf[(n0 + r) * TSTR + k + kk];
      a.y  = buf[(n0 + r) * TSTR + k + kk + 1];
      bb.x = tTc[cur][(k + kk) * 17 + r];
      bb.y = tTc[cur][(k + kk + 1) * 17 + r];
      acc = __builtin_amdgcn_wmma_f32_16x16x4_f32(
          false, a, false, bb, (short)0, acc, false, false);
    }
  }

  // overlap value chunk-0 fetch with softmax (buf0 last read 1 barrier ago)
  issue_tile(vb, 0, 0);

  {  // scatter scores: disturb heads 0,3; scale by 1/sqrt(d*H)=1/32
    const bool dist = (r == 0) || (r == 3);
#pragma unroll
    for (int j = 0; j < 8; ++j) {
      const int n = n0 + j + mh;
      const float s = dist ? 1.0f : (acc[j] + sbs[r]);
      S[r * 129 + n] = s * 0.03125f;
    }
  }
  __syncthreads();

  if (tid < NUM_HEADS) {                    // softmax per head over n<121
    const int h = tid;
    float mx = -1e30f;
    for (int n = 0; n < N_TOK; ++n) mx = fmaxf(mx, S[h * 129 + n]);
    float sum = 0.f;
    for (int n = 0; n < N_TOK; ++n) {
      const float e = __expf(S[h * 129 + n] - mx);
      S[h * 129 + n] = e;
      sum += e;
    }
    const float inv = 1.0f / sum;
    for (int n = 0; n < N_TOK; ++n) wT[n * NUM_HEADS + h] = S[h * 129 + n] * inv;
    for (int n = N_TOK; n < NPAD; ++n) wT[n * NUM_HEADS + h] = 0.f;
  }
  __syncthreads();

  // coalesced weights writeback by all 256 threads
  for (int i = tid; i < NUM_HEADS * N_TOK; i += 256) {
    const int h = i / N_TOK;
    wout[(long)b * NUM_HEADS * N_TOK + i] = wT[(i - h * N_TOK) * NUM_HEADS + h];
  }

  // ---- u pipeline: u[h,m] = sum_n w[h,n]*value[n,m], 16 chunks of 64 --------
  float* ub = u + (long)b * NUM_HEADS * MODEL_DIM;
  for (int c = 0; c < 16; ++c) {
    const int cur = c & 1;
#if USE_ASYNC_LDS
    __builtin_amdgcn_s_wait_asynccnt(0);
#else
    commit_tile(cur);
#endif
    __syncthreads();
    if (c < 15) issue_tile(vb, (c + 1) * MC, cur ^ 1);
    if (w < 4) {                            // 4 subtiles of 16 m-cols per chunk
      const float* buf = tile + cur * TILE_F;
      const int ml0 = w << 4;
      v8f au = {0.f,0.f,0.f,0.f,0.f,0.f,0.f,0.f};
#pragma unroll 8
      for (int k = 0; k < NPAD; k += 4) {   // K = token dim (zero-padded)
        v2f a, bb;
        a.x  = wT[(k + kk) * NUM_HEADS + r];
        a.y  = wT[(k + kk + 1) * NUM_HEADS + r];
        bb.x = buf[(k + kk) * TSTR + ml0 + r];
        bb.y = buf[(k + kk + 1) * TSTR + ml0 + r];
        au = __builtin_amdgcn_wmma_f32_16x16x4_f32(
            false, a, false, bb, (short)0, au, false, false);
      }
      const int mc = c * MC;
#pragma unroll
      for (int j = 0; j < 8; ++j)
        ub[(j + mh) * MODEL_DIM + mc + ml0 + r] = au[j];
    }
  }
}

// -------------------- K4: out[b, h*64+d] = u_h @ Wv_h + bv -------------------
__global__ void k_out(const float* __restrict__ u,
                      const float* __restrict__ Wv,
                      const float* __restrict__ bv,
                      float* __restrict__ outp) {
  const int wave = (blockIdx.x * blockDim.x + threadIdx.x) >> 5;
  const int lane = threadIdx.x & 31;
  const int h   = wave >> 8;
  const int rem = wave & 255;
  const int tb = (rem >> 2) << 4;
  const int td = (rem & 3) << 4;
  v8f acc = wmma_gemm_tile(u + (tb * NUM_HEADS + h) * MODEL_DIM,
                           NUM_HEADS * MODEL_DIM,
                           Wv + h * HEAD_DIM + td, MODEL_DIM, 1,
                           MODEL_DIM, lane);
  store_tile(outp + tb * MODEL_DIM + h * HEAD_DIM + td, MODEL_DIM, acc,
             bv + h * HEAD_DIM + td, lane);
}

extern "C" void kernel_launch(void* const* d_in, const int* in_sizes, int n_in,
                              void* d_out, int out_size, void* d_ws, size_t ws_size,
                              hipStream_t stream) {
  const float* query = (const float*)d_in[0];
  const float* key   = (const float*)d_in[1];
  const float* value = (const float*)d_in[2];
  const float* Wq    = (const float*)d_in[3];
  const float* bq    = (const float*)d_in[4];
  const float* Wk    = (const float*)d_in[5];
  const float* bk    = (const float*)d_in[6];
  const float* Wv    = (const float*)d_in[7];
  const float* bv    = (const float*)d_in[8];

  float* outp = (float*)d_out;                       // [1024, 1024]
  float* wout = outp + BATCH * MODEL_DIM;            // [1024, 16, 121]

  float* qproj = (float*)d_ws;                                   // 4 MB
  float* t     = qproj + BATCH * MODEL_DIM;                      // 64 MB
  float* u     = t + (size_t)BATCH * NUM_HEADS * MODEL_DIM;      // 64 MB
  float* sb    = u + (size_t)BATCH * NUM_HEADS * MODEL_DIM;      // 64 KB

  k_qproj<<<512, 256, 0, stream>>>(query, Wq, bq, qproj);
  k_sbias<<<64, 256, 0, stream>>>(qproj, bk, sb);
  k_tproj<<<8192, 256, 0, stream>>>(qproj, Wk, t);
  k_attn<<<BATCH, 256, 0, stream>>>(key, value, t, sb, u, wout);
  k_out<<<512, 256, 0, stream>>>(u, Wv, bv, outp);
}